// ProposalLayer_9552007266637
// MI455X (gfx1250) — compile-verified
//
#include <hip/hip_runtime.h>
#include <stdint.h>

#define HH 120
#define WW 120
#define KK 9
#define NB 8
#define PP (HH*WW*KK)      /* 129600 scores/boxes per batch */
#define NSORT 131072       /* next pow2 */
#define LOG2NSORT 17
#define PRE 3000
#define POST 300
#define NT 188             /* ceil(3000/16) tiles */
#define ROWU16 188         /* mask row = 188 u16 = 94 u32 */
#define ROWU32 94
#define IMGM1 1919.0f

typedef float v2f __attribute__((ext_vector_type(2)));
typedef float v8f __attribute__((ext_vector_type(8)));

#if __has_builtin(__builtin_amdgcn_wmma_f32_16x16x4_f32)
#define HAVE_WMMA4 1
#endif

// ---------------------------------------------------------------------------
// K1: regions = clip(anchors + deltas) with the reference's reinterpret-
// reshape scramble baked into the index math. Output layout (b, p, c) with
// p = (h2*W + w2)*K + k2 matching reference's final (N, HW K, 4).
// ---------------------------------------------------------------------------
__global__ void k_prep(const float* __restrict__ deltas, float* __restrict__ regions) {
    size_t tid = (size_t)blockIdx.x * blockDim.x + threadIdx.x;
    if (tid >= (size_t)NB * PP * 4) return;
    int b = (int)(tid / ((size_t)PP * 4));
    int o = (int)(tid % ((size_t)PP * 4));
    int c2 = o & 3;
    int k2 = (o >> 2) % KK;
    int hw = (o >> 2) / KK;
    int h2 = hw / WW, w2 = hw % WW;
    // position t inside the original (H,W,K,4)-flat array
    int t = ((k2 * 4 + c2) * HH + h2) * WW + w2;
    int h  = t / (WW * KK * 4);
    int w  = (t / (KK * 4)) % WW;
    int k  = (t % (KK * 4)) / 4;
    int cc = t & 3;
    float ratio = (k / 3 == 0) ? 0.5f : ((k / 3 == 1) ? 1.0f : 2.0f);
    float scale = (k % 3 == 0) ? 8.0f : ((k % 3 == 1) ? 16.0f : 32.0f);
    float sq  = sqrtf(ratio);
    float wsv = 16.0f * scale / sq;
    float hsv = 16.0f * scale * sq;
    float cx = (w + 0.5f) * 16.0f, cy = (h + 0.5f) * 16.0f;
    float av;
    if      (cc == 0) av = cx - 0.5f * wsv;
    else if (cc == 1) av = cy - 0.5f * hsv;
    else if (cc == 2) av = cx + 0.5f * wsv;
    else              av = cy + 0.5f * hsv;
    float d = deltas[(((size_t)b * 36 + (k * 4 + cc)) * HH + h) * WW + w];
    float v = fminf(fmaxf(av + d, 0.0f), IMGM1);
    regions[(size_t)b * (PP * 4) + o] = v;
}

// ---------------------------------------------------------------------------
// K2: build 64-bit sort items: key = (~orderFlip(scoreBits) << 32) | index.
// Ascending sort => score descending, index ascending (top_k tie-break).
// ---------------------------------------------------------------------------
__global__ void k_keys(const float* __restrict__ cls, uint64_t* __restrict__ sb) {
    size_t tid = (size_t)blockIdx.x * blockDim.x + threadIdx.x;
    if (tid >= (size_t)NB * NSORT) return;
    int b = (int)(tid >> LOG2NSORT);
    int s = (int)(tid & (NSORT - 1));
    uint64_t item;
    if (s < PP) {
        uint32_t u = __float_as_uint(cls[(size_t)b * PP + s]);
        uint32_t f = u ^ ((u >> 31) ? 0xFFFFFFFFu : 0x80000000u);
        item = ((uint64_t)(~f) << 32) | (uint32_t)s;
    } else {
        item = ~0ULL;   // pad sorts last
    }
    sb[tid] = item;
}

__global__ void k_bitonic(uint64_t* __restrict__ sb, unsigned j, unsigned kk) {
    size_t tid = (size_t)blockIdx.x * blockDim.x + threadIdx.x;
    unsigned i = (unsigned)(tid & (NSORT - 1));
    uint64_t* base = sb + (tid - i);
    unsigned ixj = i ^ j;
    if (ixj > i) {
        uint64_t a = base[i], c = base[ixj];
        bool up = ((i & kk) == 0);
        if (up ? (a > c) : (a < c)) { base[i] = c; base[ixj] = a; }
    }
}

// ---------------------------------------------------------------------------
// K3: gather top-3000 boxes + scores per batch.
// ---------------------------------------------------------------------------
__global__ void k_gather(const uint64_t* __restrict__ sb, const float* __restrict__ cls,
                         const float* __restrict__ regions,
                         float* __restrict__ tb, float* __restrict__ ts) {
    int tid = blockIdx.x * blockDim.x + threadIdx.x;
    if (tid >= NB * PRE) return;
    int b = tid / PRE, r = tid % PRE;
    uint64_t it = sb[((size_t)b << LOG2NSORT) + r];
    uint32_t idx = (uint32_t)it;
    ts[tid] = cls[(size_t)b * PP + idx];
    const float* rp = regions + ((size_t)b * PP + idx) * 4;
    float* op = tb + (size_t)tid * 4;
    op[0] = rp[0]; op[1] = rp[1]; op[2] = rp[2]; op[3] = rp[3];
}

__global__ void k_zero32(uint32_t* __restrict__ p, size_t n) {
    size_t tid = (size_t)blockIdx.x * blockDim.x + threadIdx.x;
    if (tid < n) p[tid] = 0;
}

// ---------------------------------------------------------------------------
// K4: NMS suppression bitmask. One wave32 per (batch, ti, tj>=ti) 16x16 tile.
// Boxes staged to LDS via CDNA5 GLOBAL_LOAD_ASYNC_TO_LDS_B128; the pairwise
// union base (areas[i]+areas[j]) computed via V_WMMA_F32_16X16X4_F32 outer
// product; suppression bits gathered cross-lane with ballot, stored as u16
// (row-exclusive, no atomics).
// ---------------------------------------------------------------------------
__global__ void __launch_bounds__(32) k_mask(const float* __restrict__ tb,
                                             unsigned short* __restrict__ m16) {
    int tj = blockIdx.x, ti = blockIdx.y, b = blockIdx.z;
    if (tj < ti) return;                       // uniform: EXEC stays all-ones
    int lane = (int)threadIdx.x;
    __shared__ float bx[2 * 16 * 4];           // [0:16) = I-tile, [16:32) = J-tile

    int sj   = lane & 15;
    int tile = (lane < 16) ? ti : tj;
    int row  = tile * 16 + sj;
    if (row > PRE - 1) row = PRE - 1;          // clamp (pred excludes >=PRE)

    // --- async stage: each lane pulls one float4 box into its LDS slot ---
    uint32_t ldsaddr = (uint32_t)(uintptr_t)&bx[lane * 4];
    uint32_t voff    = (uint32_t)row * 16u;
    uint64_t sbase   = (uint64_t)(uintptr_t)(tb + (size_t)b * PRE * 4);
    asm volatile("global_load_async_to_lds_b128 %0, %1, %2"
                 :: "v"(ldsaddr), "v"(voff), "s"(sbase) : "memory");
#if __has_builtin(__builtin_amdgcn_s_wait_asynccnt)
    __builtin_amdgcn_s_wait_asynccnt(0);
#else
    asm volatile("s_wait_asynccnt 0x0" ::: "memory");
#endif

    volatile float* bxv = bx;
    float bjx = bxv[(16 + sj) * 4 + 0], bjy = bxv[(16 + sj) * 4 + 1];
    float bjz = bxv[(16 + sj) * 4 + 2], bjw = bxv[(16 + sj) * 4 + 3];
    float aix = bxv[sj * 4 + 0], aiy = bxv[sj * 4 + 1];
    float aiz = bxv[sj * 4 + 2], aiw = bxv[sj * 4 + 3];
    float areaJ = (bjz - bjx + 1.0f) * (bjw - bjy + 1.0f);
    float areaI = (aiz - aix + 1.0f) * (aiw - aiy + 1.0f);
    bool hi = lane >= 16;

#ifdef HAVE_WMMA4
    // A (16x4): row M = [areaI[M], 1, 0, 0] ; B (4x16): col N = [1, areaJ[N], 0, 0]
    // => D[M][N] = areaI[M] + areaJ[N]. Lanes 16-31 carry K=2,3 rows (zeros).
    v2f A, Bm;
    A.x  = hi ? 0.0f : areaI;  A.y  = hi ? 0.0f : 1.0f;
    Bm.x = hi ? 0.0f : 1.0f;   Bm.y = hi ? 0.0f : areaJ;
    v8f C = {};
    v8f D = __builtin_amdgcn_wmma_f32_16x16x4_f32(false, A, false, Bm,
                                                  (short)0, C, false, false);
#endif

    int jGlob = tj * 16 + sj;
    for (int v = 0; v < 8; v++) {
        int iLoc  = v + (hi ? 8 : 0);
        int iGlob = ti * 16 + iLoc;
        float bx0 = bxv[iLoc * 4 + 0], by0 = bxv[iLoc * 4 + 1];
        float bx1 = bxv[iLoc * 4 + 2], by1 = bxv[iLoc * 4 + 3];
        float xx1 = fmaxf(bx0, bjx), yy1 = fmaxf(by0, bjy);
        float xx2 = fminf(bx1, bjz), yy2 = fminf(by1, bjw);
        float iw = fmaxf(xx2 - xx1 + 1.0f, 0.0f);
        float ih = fmaxf(yy2 - yy1 + 1.0f, 0.0f);
        float inter = iw * ih;
#ifdef HAVE_WMMA4
        float sum = D[v];                      // areaI[M] + areaJ[N] via WMMA
#else
        float sum = (bx1 - bx0 + 1.0f) * (by1 - by0 + 1.0f) + areaJ;
#endif
        float iou = inter / (sum - inter);
        bool pred = (iou > 0.5f) & (jGlob > iGlob) & (jGlob < PRE) & (iGlob < PRE);
        unsigned long long bal = __ballot(pred); // lo16: row v, hi16: row v+8
        if (lane == 0) {
            int r0 = ti * 16 + v;
            if (r0 < PRE)
                m16[((size_t)b * PRE + r0) * ROWU16 + tj] = (unsigned short)(bal & 0xFFFFu);
            int r1 = r0 + 8;
            if (r1 < PRE)
                m16[((size_t)b * PRE + r1) * ROWU16 + tj] = (unsigned short)((bal >> 16) & 0xFFFFu);
        }
    }
}

// ---------------------------------------------------------------------------
// K5: sequential greedy reduction, one wave32 per batch. The 94-word removed
// bitset lives in 3 VGPRs/lane; cross-lane reads via shfl (no barriers).
// Writes the first 300 kept (boxes+scores); slots beyond kept-count are 0.
// ---------------------------------------------------------------------------
__global__ void __launch_bounds__(32) k_reduce(const unsigned* __restrict__ m32,
                                               const float* __restrict__ tb,
                                               const float* __restrict__ ts,
                                               float* __restrict__ selb,
                                               float* __restrict__ sels) {
    int b = blockIdx.x;
    int lane = (int)threadIdx.x;
    for (int t = lane; t < POST * 4; t += 32) selb[(size_t)b * POST * 4 + t] = 0.0f;
    for (int t = lane; t < POST;     t += 32) sels[(size_t)b * POST + t]     = 0.0f;

    unsigned r0 = 0, r1 = 0, r2 = 0;           // lane owns words [3*lane, 3*lane+2]
    int w0 = lane * 3;
    int cnt = 0;
    for (int i = 0; i < PRE && cnt < POST; i++) {
        int wi = i >> 5;                       // uniform
        int slot = wi % 3, owner = wi / 3;     // uniform
        unsigned rw = (slot == 0) ? r0 : ((slot == 1) ? r1 : r2);
        rw = (unsigned)__shfl((int)rw, owner, 32);
        if (!((rw >> (i & 31)) & 1u)) {        // kept
            const unsigned* rowp = m32 + ((size_t)b * PRE + i) * ROWU32;
            __builtin_prefetch(rowp + ROWU32, 0, 0);
            if (w0     < ROWU32) r0 |= rowp[w0];
            if (w0 + 1 < ROWU32) r1 |= rowp[w0 + 1];
            if (w0 + 2 < ROWU32) r2 |= rowp[w0 + 2];
            if (lane == 0) sels[(size_t)b * POST + cnt] = ts[(size_t)b * PRE + i];
            if (lane < 4)
                selb[((size_t)b * POST + cnt) * 4 + lane] = tb[((size_t)b * PRE + i) * 4 + lane];
            cnt++;
        }
    }
}

// ---------------------------------------------------------------------------
// K6: assemble output (N,300,5): col0 = batch (N-1)'s scores (reference quirk),
// cols 1..4 = this batch's boxes.
// ---------------------------------------------------------------------------
__global__ void k_out(const float* __restrict__ selb, const float* __restrict__ sels,
                      float* __restrict__ out) {
    int tid = blockIdx.x * blockDim.x + threadIdx.x;
    if (tid >= NB * POST) return;
    int r = tid % POST;
    float* o = out + (size_t)tid * 5;
    o[0] = sels[(size_t)(NB - 1) * POST + r];
    const float* bp = selb + (size_t)tid * 4;
    o[1] = bp[0]; o[2] = bp[1]; o[3] = bp[2]; o[4] = bp[3];
}

// ---------------------------------------------------------------------------
static inline size_t alignUp(size_t x) { return (x + 255) & ~(size_t)255; }

extern "C" void kernel_launch(void* const* d_in, const int* in_sizes, int n_in,
                              void* d_out, int out_size, void* d_ws, size_t ws_size,
                              hipStream_t stream) {
    const float* cls    = (const float*)d_in[0];   // (8,9,120,120)
    const float* deltas = (const float*)d_in[1];   // (8,36,120,120)
    float* out = (float*)d_out;                    // (8,300,5)
    char* ws = (char*)d_ws;

    size_t oSort = 0;
    size_t oReg  = oSort + alignUp((size_t)NB * NSORT * 8);        // 8.39 MB
    size_t oTB   = oReg  + alignUp((size_t)NB * PP * 4 * 4);       // 16.59 MB
    size_t oTS   = oTB   + alignUp((size_t)NB * PRE * 4 * 4);
    size_t oMask = oTS   + alignUp((size_t)NB * PRE * 4);
    size_t oSelB = oMask + alignUp((size_t)NB * PRE * ROWU16 * 2); // 9.02 MB
    size_t oSelS = oSelB + alignUp((size_t)NB * POST * 4 * 4);
    (void)ws_size; (void)in_sizes; (void)n_in; (void)out_size;     // ~34.6 MB total

    uint64_t*       sortbuf = (uint64_t*)(ws + oSort);
    float*          regions = (float*)(ws + oReg);
    float*          tb      = (float*)(ws + oTB);
    float*          tsc     = (float*)(ws + oTS);
    unsigned short* m16     = (unsigned short*)(ws + oMask);
    unsigned*       m32     = (unsigned*)(ws + oMask);
    float*          selb    = (float*)(ws + oSelB);
    float*          sels    = (float*)(ws + oSelS);

    k_prep<<<(NB * PP * 4 + 255) / 256, 256, 0, stream>>>(deltas, regions);
    k_keys<<<(NB * NSORT) / 256, 256, 0, stream>>>(cls, sortbuf);
    for (unsigned kk = 2; kk <= NSORT; kk <<= 1)
        for (unsigned jj = kk >> 1; jj > 0; jj >>= 1)
            k_bitonic<<<(NB * NSORT) / 256, 256, 0, stream>>>(sortbuf, jj, kk);
    k_gather<<<(NB * PRE + 255) / 256, 256, 0, stream>>>(sortbuf, cls, regions, tb, tsc);
    size_t maskWords = (size_t)NB * PRE * ROWU32;
    k_zero32<<<(unsigned)((maskWords + 255) / 256), 256, 0, stream>>>(m32, maskWords);
    k_mask<<<dim3(NT, NT, NB), 32, 0, stream>>>(tb, m16);
    k_reduce<<<NB, 32, 0, stream>>>(m32, tb, tsc, selb, sels);
    k_out<<<(NB * POST + 255) / 256, 256, 0, stream>>>(selb, sels, out);
}